// ROIFeatureExtraction_33921651704214
// MI455X (gfx1250) — compile-verified
//
#include <hip/hip_runtime.h>
#include <hip/hip_bf16.h>
#include <stdint.h>

#define NA     3584      // total anchors per batch (512+1024+2048)
#define TOPK_N 128
#define NBIN   196       // 14*14
#define NMSTHR 0.7f
#define ROI_CHUNK 32

typedef unsigned int v4u  __attribute__((ext_vector_type(4)));
typedef int          v8i_ __attribute__((ext_vector_type(8)));
typedef int          v4i_ __attribute__((ext_vector_type(4)));
typedef float        v2f  __attribute__((ext_vector_type(2)));
typedef float        v8f  __attribute__((ext_vector_type(8)));

#if defined(__AMDGCN__) && __has_builtin(__builtin_amdgcn_tensor_load_to_lds)
#define HAVE_TDM 1
#endif
#if defined(__AMDGCN__) && __has_builtin(__builtin_amdgcn_wmma_f32_16x16x4_f32)
#define HAVE_WMMA 1
#endif

__device__ __forceinline__ int   d_min(int a, int b)   { return a < b ? a : b; }
__device__ __forceinline__ float d_clampf(float v, float lo, float hi) {
  return fminf(fmaxf(v, lo), hi);
}

// ---------------------------------------------------------------------------
// TDM: stage nElems f32 (contiguous) from global into LDS. Call from wave 0
// only (TDM issues per-wave, EXEC ignored). Descriptor per ISA 08 §8.3-8.4:
// one row of nElems 4-byte elements, type=2, count=1.
// ---------------------------------------------------------------------------
#ifdef HAVE_TDM
__device__ __forceinline__ void tdm_load_tile(const float* src, float* ldsDst,
                                              uint32_t nElems) {
  uint64_t ga  = (uint64_t)(uintptr_t)src;
  uint32_t lds = (uint32_t)(uintptr_t)ldsDst;     // low 32 bits = LDS byte offset
  v4u g0;
  g0[0] = 1u;                                     // count=1, no gather, user mode
  g0[1] = lds;                                    // lds_addr
  g0[2] = (uint32_t)ga;                           // global_addr[31:0]
  g0[3] = ((uint32_t)(ga >> 32) & 0x01FFFFFFu)    // global_addr[56:32]
          | (2u << 30);                           // type = 2 (image)
  const uint32_t N = nElems;
  v8i_ g1;
  g1[0] = (int)(2u << 16);                        // data_size = 4 bytes
  g1[1] = (int)((N & 0xFFFFu) << 16);             // tensor_dim0[15:0]
  g1[2] = (int)((N >> 16) | (1u << 16));          // tensor_dim0[31:16], tensor_dim1=1
  g1[3] = (int)((N & 0xFFFFu) << 16);             // tile_dim0 = N (<=65535)
  g1[4] = 1;                                      // tile_dim1=1, tile_dim2=0
  g1[5] = (int)N;                                 // tensor_dim0_stride lo32
  g1[6] = (int)((N & 0xFFFFu) << 16);             // stride0 hi=0, stride1[15:0]
  g1[7] = (int)(N >> 16);                         // stride1[47:16]
  v4i_ z4 = (v4i_)0;
  v8i_ z8 = (v8i_)0;
  __builtin_amdgcn_tensor_load_to_lds(g0, g1, z4, z4, z8, 0);
  __builtin_amdgcn_s_wait_tensorcnt(0);
}
#endif

// ---------------------------------------------------------------------------
// K0: per-anchor prep: level channel, stride-scaled box (cxcywh), xyxy box
// ---------------------------------------------------------------------------
__global__ void prep_kernel(const float* __restrict__ a32,
                            const float* __restrict__ a16,
                            const float* __restrict__ a8,
                            float* __restrict__ wsBoxes,
                            float* __restrict__ wsXyxy,
                            int*   __restrict__ wsLev,
                            int B) {
  int g = blockIdx.x * blockDim.x + threadIdx.x;
  if (g >= B * NA) return;
  int b = g / NA, j = g - b * NA;
  const float* a; float st;
  if (j < 512)        { a = a32 + ((size_t)b * 512  + j)          * 4; st = 32.f; }
  else if (j < 1536)  { a = a16 + ((size_t)b * 1024 + (j - 512))  * 4; st = 16.f; }
  else                { a = a8  + ((size_t)b * 2048 + (j - 1536)) * 4; st =  8.f; }
  float cx = a[0], cy = a[1], w = a[2], h = a[3];
  float s  = sqrtf(w * h);
  float lv = floorf(3.0f + log2f(s * (1.0f / 224.0f)));
  lv = fminf(fmaxf(lv, 1.0f), 4.0f);
  wsLev[g] = (int)lv;
  float bx = cx * st, by = cy * st, bw = w * st, bh = h * st;
  float* bo = wsBoxes + (size_t)g * 4;
  bo[0] = bx; bo[1] = by; bo[2] = bw; bo[3] = bh;
  float* xo = wsXyxy + (size_t)g * 4;
  xo[0] = bx - bw * 0.5f; xo[1] = by - bh * 0.5f;
  xo[2] = bx + bw * 0.5f; xo[3] = by + bh * 0.5f;
}

// ---------------------------------------------------------------------------
// K1: per (batch, level): bitonic sort by masked score, greedy NMS,
//     compact first 128 kept finite -> rois (cxcywh) + valid flag
// ---------------------------------------------------------------------------
__global__ __launch_bounds__(1024)
void nms_topk_kernel(const float* __restrict__ s32,
                     const float* __restrict__ s16,
                     const float* __restrict__ s8,
                     const float* __restrict__ wsBoxes,
                     const float* __restrict__ wsXyxy,
                     const int*   __restrict__ wsLev,
                     float* __restrict__ wsRois,
                     float* __restrict__ wsValid) {
  const int nl  = blockIdx.x + 1;   // level channel 1..3
  const int b   = blockIdx.y;
  const int tid = threadIdx.x;
  const float NEG = -__builtin_inff();

  __shared__ float skey[4096];
  __shared__ int   ssid[4096];
  __shared__ unsigned char ssup[NA];
  __shared__ int   ssel[TOPK_N];
  __shared__ int   scnt;

  for (int i = tid; i < 4096; i += 1024) {
    float v = NEG;
    if (i < NA) {
      float sc = (i < 512)  ? s32[(size_t)b * 512 + i]
               : (i < 1536) ? s16[(size_t)b * 1024 + (i - 512)]
                            : s8 [(size_t)b * 2048 + (i - 1536)];
      v = (wsLev[(size_t)b * NA + i] == nl) ? sc : NEG;
    }
    skey[i] = v;
    ssid[i] = i;
  }
  for (int i = tid; i < NA; i += 1024) ssup[i] = 0;
  __syncthreads();

  // bitonic sort, descending
  for (int k = 2; k <= 4096; k <<= 1) {
    for (int j2 = k >> 1; j2 > 0; j2 >>= 1) {
      for (int i = tid; i < 4096; i += 1024) {
        int p = i ^ j2;
        if (p > i) {
          bool up = ((i & k) == 0);
          float ki = skey[i], kp = skey[p];
          bool sw = up ? (ki < kp) : (ki > kp);
          if (sw) {
            skey[i] = kp; skey[p] = ki;
            int t = ssid[i]; ssid[i] = ssid[p]; ssid[p] = t;
          }
        }
      }
      __syncthreads();
    }
  }

  // greedy NMS over sorted order
  const float* xy = wsXyxy + (size_t)b * NA * 4;
  for (int i = 0; i < NA; ++i) {
    float ki = skey[i];
    if (ki == NEG) break;            // uniform across block
    if (!ssup[i]) {
      int aidx = ssid[i];
      float ax1 = xy[aidx * 4 + 0], ay1 = xy[aidx * 4 + 1];
      float ax2 = xy[aidx * 4 + 2], ay2 = xy[aidx * 4 + 3];
      float aar = fmaxf(ax2 - ax1, 0.f) * fmaxf(ay2 - ay1, 0.f);
      for (int q = i + 1 + tid; q < NA; q += 1024) {
        if (ssup[q]) continue;
        if (skey[q] == NEG) continue;
        int cidx = ssid[q];
        float cx1 = xy[cidx * 4 + 0], cy1 = xy[cidx * 4 + 1];
        float cx2 = xy[cidx * 4 + 2], cy2 = xy[cidx * 4 + 3];
        float car = fmaxf(cx2 - cx1, 0.f) * fmaxf(cy2 - cy1, 0.f);
        float ix1 = fmaxf(ax1, cx1), iy1 = fmaxf(ay1, cy1);
        float ix2 = fminf(ax2, cx2), iy2 = fminf(ay2, cy2);
        float inter = fmaxf(ix2 - ix1, 0.f) * fmaxf(iy2 - iy1, 0.f);
        float uni = aar + car - inter;
        float iou = inter / fmaxf(uni, 1e-9f);
        if (iou > NMSTHR) ssup[q] = 1;
      }
    }
    __syncthreads();
  }

  if (tid == 0) {
    int cnt = 0;
    for (int i = 0; i < NA && cnt < TOPK_N; ++i)
      if (skey[i] != NEG && !ssup[i]) ssel[cnt++] = ssid[i];
    scnt = cnt;
  }
  __syncthreads();

  if (tid < TOPK_N) {
    size_t ro = ((size_t)b * 3 + (nl - 1)) * TOPK_N + tid;
    if (tid < scnt) {
      int aidx = ssel[tid];
      const float* bo = wsBoxes + ((size_t)b * NA + aidx) * 4;
      wsRois[ro * 4 + 0] = bo[0];
      wsRois[ro * 4 + 1] = bo[1];
      wsRois[ro * 4 + 2] = bo[2];
      wsRois[ro * 4 + 3] = bo[3];
      wsValid[ro] = 1.0f;
    } else {
      wsRois[ro * 4 + 0] = 0.f; wsRois[ro * 4 + 1] = 0.f;
      wsRois[ro * 4 + 2] = 0.f; wsRois[ro * 4 + 3] = 0.f;
      wsValid[ro] = 0.0f;
    }
  }
}

// ---------------------------------------------------------------------------
// shared geometry helper: 28 separable taps per axis for one roi
// ---------------------------------------------------------------------------
__device__ __forceinline__ void roi_geometry(int tid, const float* rp, int H, int W,
                                             int* sx0, int* sx1, float* slx, float* svx,
                                             int* sy0, int* sy1, float* sly, float* svy) {
  if (tid < 56) {
    const int  j28 = tid < 28 ? tid : tid - 28;
    const bool isY = tid >= 28;
    float lo  = isY ? rp[1] : rp[0];
    float hi  = isY ? rp[3] : rp[2];
    float dim = isY ? (float)H : (float)W;
    float rr  = fmaxf(hi - lo, 1.0f);
    float bsz = rr * (1.0f / 14.0f);
    float fj  = (float)(j28 >> 1) + 0.25f + 0.5f * (float)(j28 & 1);
    float s   = lo + bsz * fj;
    float vv  = (s > -1.0f && s < dim) ? 1.0f : 0.0f;
    float sc  = d_clampf(s, 0.0f, dim - 1.0f);
    float f0  = floorf(sc);
    int   i0  = (int)f0;
    int   i1  = d_min(i0 + 1, (int)dim - 1);
    float l   = sc - f0;
    if (isY) { sy0[j28] = i0; sy1[j28] = i1; sly[j28] = l; svy[j28] = vv; }
    else     { sx0[j28] = i0; sx1[j28] = i1; slx[j28] = l; svx[j28] = vv; }
  }
}

// ---------------------------------------------------------------------------
// K2a: ROI-align for the 32x32 level with WMMA pooling.
// Workgroup = (16-channel tile, roi chunk, batch). TDM stages the 64KB channel
// tile into LDS; bilinear samples (16 x 28 x 28) go to an LDS sample buffer;
// the 2x2 sample averaging is lowered to two GEMM chains on
// V_WMMA_F32_16X16X4_F32:  Out = PY(14x28) . S . PX(28x14), PX/PY = 0.5-pairing.
// Fragment layouts per ISA 05 §7.12.2 (f32 16x4 A/B striping, 16x16 C/D).
// ---------------------------------------------------------------------------
#define K2A_HW   1024
#define K2A_SST  785     // sample buf channel stride (784 + 1 pad)
#define K2A_TST  449     // T buf channel stride (28*16 + 1 pad)

__global__ __launch_bounds__(256)
void roi_align_lev0_kernel(const float* __restrict__ p32,
                           const float* __restrict__ wsRois,
                           const float* __restrict__ wsValid,
                           float* __restrict__ out) {
  extern __shared__ unsigned char smem[];
  float* sfeat = (float*)smem;                    // 16 * 1024
  float* Sbuf  = sfeat + 16 * K2A_HW;             // 16 * 785
  float* Tbuf  = Sbuf + 16 * K2A_SST;             // 16 * 449
  int*   sx0   = (int*)(Tbuf + 16 * K2A_TST);
  int*   sx1   = sx0 + 28;
  int*   sy0   = sx1 + 28;
  int*   sy1   = sy0 + 28;
  float* slx   = (float*)(sy1 + 28);
  float* sly   = slx + 28;
  float* svx   = sly + 28;
  float* svy   = svx + 28;

  const int ctile = blockIdx.x;        // 16 tiles of 16 channels
  const int chunk = blockIdx.y;
  const int b     = blockIdx.z;
  const int tid   = threadIdx.x;
  const int lane  = tid & 31;
  const int wave  = tid >> 5;
  const bool hif  = lane >= 16;
  const int  lm   = lane & 15;

  const float* src = p32 + ((size_t)b * 256 + (size_t)ctile * 16) * K2A_HW;
#ifdef HAVE_TDM
  __builtin_prefetch(src, 0, 0);
  if (tid < 32) tdm_load_tile(src, sfeat, 16 * K2A_HW);
#else
  for (int i = tid; i < 16 * K2A_HW; i += 256) sfeat[i] = src[i];
#endif
  __syncthreads();

  const size_t lvbase = (size_t)b * 3 * TOPK_N;   // level 0
  for (int r = 0; r < ROI_CHUNK; ++r) {
    const int roi = chunk * ROI_CHUNK + r;
    const float* rp = wsRois + (lvbase + roi) * 4;
    const float vroi = wsValid[lvbase + roi];

    __syncthreads();
    roi_geometry(tid, rp, 32, 32, sx0, sx1, slx, svx, sy0, sy1, sly, svy);
    __syncthreads();

    // all 16 x 784 bilinear samples -> Sbuf (no 0.25, folded into PX.PY)
    for (int idx = tid; idx < 16 * 784; idx += 256) {
      int c  = idx / 784;
      int p  = idx - c * 784;
      int py = p / 28, px = p - py * 28;
      const float* fb = sfeat + c * K2A_HW;
      const int y0 = sy0[py], y1 = sy1[py];
      const float ly = sly[py], hy = 1.0f - ly;
      const int x0 = sx0[px], x1 = sx1[px];
      const float lx = slx[px], hx = 1.0f - lx;
      const float* r0 = fb + y0 * 32;
      const float* r1 = fb + y1 * 32;
      float val = r0[x0] * (hy * hx) + r0[x1] * (hy * lx)
                + r1[x0] * (ly * hx) + r1[x1] * (ly * lx);
      Sbuf[c * K2A_SST + p] = val * (svy[py] * svx[px]);
    }
    __syncthreads();

#ifdef HAVE_WMMA
    // stage 1: T[c, py, bx] = sum_px S[c, py, px] * PX[px, bx], PX = 0.5 pairing
    for (int py = wave; py < 28; py += 8) {
      v8f acc = {};
      for (int kt = 0; kt < 7; ++kt) {
        const int ka = kt * 4 + (hif ? 2 : 0);    // A/B VGPR0 K, VGPR1 = K+1
        v2f a, bb;
        a[0]  = Sbuf[lm * K2A_SST + py * 28 + ka];
        a[1]  = Sbuf[lm * K2A_SST + py * 28 + ka + 1];
        bb[0] = ((ka >> 1)       == lm) ? 0.5f : 0.0f;
        bb[1] = (((ka + 1) >> 1) == lm) ? 0.5f : 0.0f;
        acc = __builtin_amdgcn_wmma_f32_16x16x4_f32(false, a, false, bb,
                                                    (short)0, acc, false, false);
      }
#pragma unroll
      for (int v = 0; v < 8; ++v) {               // D: VGPR v -> row v / v+8
        int c = v + (hif ? 8 : 0);
        Tbuf[c * K2A_TST + py * 16 + lm] = acc[v];
      }
    }
    __syncthreads();
    // stage 2: Out[c, by, bx] = sum_py PY[by, py] * T[c, py, bx]
    for (int bx = wave; bx < 14; bx += 8) {
      v8f acc = {};
      for (int kt = 0; kt < 7; ++kt) {
        const int ka = kt * 4 + (hif ? 2 : 0);
        v2f a, bb;
        a[0]  = Tbuf[lm * K2A_TST + ka * 16 + bx];
        a[1]  = Tbuf[lm * K2A_TST + (ka + 1) * 16 + bx];
        bb[0] = ((ka >> 1)       == lm) ? 0.5f : 0.0f;
        bb[1] = (((ka + 1) >> 1) == lm) ? 0.5f : 0.0f;
        acc = __builtin_amdgcn_wmma_f32_16x16x4_f32(false, a, false, bb,
                                                    (short)0, acc, false, false);
      }
      const int by = lm;
#pragma unroll
      for (int v = 0; v < 8; ++v) {
        int c = v + (hif ? 8 : 0);
        if (by < 14) {
          const size_t oidx =
              ((((size_t)b * 384 + roi) * 256 + (size_t)ctile * 16 + c) * NBIN)
              + by * 14 + bx;
          out[oidx] = acc[v] * vroi;
        }
      }
    }
#else
    // scalar pooling fallback
    for (int idx = tid; idx < 16 * NBIN; idx += 256) {
      int c = idx / NBIN, bin = idx - c * NBIN;
      int by = bin / 14, bx = bin - by * 14;
      const float* Sc = Sbuf + c * K2A_SST;
      float o = 0.25f * (Sc[(2 * by) * 28 + 2 * bx] + Sc[(2 * by) * 28 + 2 * bx + 1]
                       + Sc[(2 * by + 1) * 28 + 2 * bx] + Sc[(2 * by + 1) * 28 + 2 * bx + 1]);
      const size_t oidx =
          ((((size_t)b * 384 + roi) * 256 + (size_t)ctile * 16 + c) * NBIN) + bin;
      out[oidx] = o * vroi;
    }
#endif
  }
}

// ---------------------------------------------------------------------------
// K2b: ROI-align for the 64x64 and 128x128 levels (TDM 128KB tile + scalar
// 4-sample average; <16 channels per tile so no WMMA grouping possible).
// ---------------------------------------------------------------------------
#define TILE_ELEMS 32768
#define K2_BLK 256

__global__ __launch_bounds__(K2_BLK)
void roi_align_kernel(const float* __restrict__ p16,
                      const float* __restrict__ p8,
                      const float* __restrict__ wsRois,
                      const float* __restrict__ wsValid,
                      float* __restrict__ out) {
  extern __shared__ unsigned char smem[];
  float* sfeat = (float*)smem;                                   // 131072 B
  int*   sx0   = (int*)  (smem + TILE_ELEMS * 4);
  int*   sx1   = sx0 + 28;
  int*   sy0   = sx1 + 28;
  int*   sy1   = sy0 + 28;
  float* slx   = (float*)(sy1 + 28);
  float* sly   = slx + 28;
  float* svx   = sly + 28;
  float* svy   = svx + 28;

  const int job   = blockIdx.x;     // 0..159: flattened (level 1..2, tile)
  const int chunk = blockIdx.y;
  const int b     = blockIdx.z;
  const int tid   = threadIdx.x;

  int lev, ctile, TC, H, W;
  const float* feat;
  if (job < 32) { lev = 1; ctile = job;      TC = 8; H = 64;  W = 64;  feat = p16; }
  else          { lev = 2; ctile = job - 32; TC = 2; H = 128; W = 128; feat = p8;  }
  const int HW = H * W;
  const float* src = feat + ((size_t)b * 256 + (size_t)ctile * TC) * HW;

#ifdef HAVE_TDM
  __builtin_prefetch(src, 0, 0);
  if (tid < 32) tdm_load_tile(src, sfeat, TILE_ELEMS);
#else
  for (int i = tid; i < TILE_ELEMS; i += K2_BLK) sfeat[i] = src[i];
#endif
  __syncthreads();

  const size_t lvbase = ((size_t)b * 3 + lev) * TOPK_N;
  const int total = TC * NBIN;

  for (int r = 0; r < ROI_CHUNK; ++r) {
    const int roi = chunk * ROI_CHUNK + r;
    const float* rp = wsRois + (lvbase + roi) * 4;
    const float vroi = wsValid[lvbase + roi];

    __syncthreads();
    roi_geometry(tid, rp, H, W, sx0, sx1, slx, svx, sy0, sy1, sly, svy);
    __syncthreads();

    for (int idx = tid; idx < total; idx += K2_BLK) {
      const int c   = idx / NBIN;
      const int bin = idx - c * NBIN;
      const int by  = bin / 14;
      const int bx  = bin - by * 14;
      const float* fb = sfeat + c * HW;
      float acc = 0.f;
#pragma unroll
      for (int sy = 0; sy < 2; ++sy) {
        const int py = 2 * by + sy;
        const int y0 = sy0[py], y1 = sy1[py];
        const float ly = sly[py], vy = svy[py];
        const float hy = 1.0f - ly;
        const float* r0 = fb + y0 * W;
        const float* r1 = fb + y1 * W;
#pragma unroll
        for (int sx = 0; sx < 2; ++sx) {
          const int px = 2 * bx + sx;
          const int x0 = sx0[px], x1 = sx1[px];
          const float lx = slx[px], vx = svx[px];
          const float hx = 1.0f - lx;
          float val = r0[x0] * (hy * hx) + r0[x1] * (hy * lx)
                    + r1[x0] * (ly * hx) + r1[x1] * (ly * lx);
          acc += val * (vy * vx);
        }
      }
      const size_t oidx =
          ((((size_t)b * 384 + (size_t)lev * TOPK_N + roi) * 256
            + (size_t)ctile * TC + c) * NBIN) + bin;
      out[oidx] = acc * 0.25f * vroi;
    }
  }
}

// ---------------------------------------------------------------------------
// Launch
// ---------------------------------------------------------------------------
extern "C" void kernel_launch(void* const* d_in, const int* in_sizes, int n_in,
                              void* d_out, int out_size, void* d_ws, size_t ws_size,
                              hipStream_t stream) {
  const float* p32 = (const float*)d_in[0];
  const float* p16 = (const float*)d_in[1];
  const float* p8  = (const float*)d_in[2];
  // d_in[3] (p4) unused by the reference
  const float* a32 = (const float*)d_in[4];
  const float* a16 = (const float*)d_in[5];
  const float* a8  = (const float*)d_in[6];
  const float* s32 = (const float*)d_in[7];
  const float* s16 = (const float*)d_in[8];
  const float* s8  = (const float*)d_in[9];
  float* out = (float*)d_out;
  (void)n_in; (void)out_size; (void)ws_size;

  const int B = in_sizes[7] / 512;   // scores_32 is (B, 512)

  // workspace layout (floats)
  float* wsBoxes = (float*)d_ws;                       // B*NA*4
  float* wsXyxy  = wsBoxes + (size_t)B * NA * 4;       // B*NA*4
  int*   wsLev   = (int*)(wsXyxy + (size_t)B * NA * 4);// B*NA
  float* wsRois  = (float*)(wsLev + (size_t)B * NA);   // B*3*128*4
  float* wsValid = wsRois + (size_t)B * 3 * TOPK_N * 4;// B*3*128

  // K0: prep
  {
    int total = B * NA;
    int blocks = (total + 255) / 256;
    prep_kernel<<<blocks, 256, 0, stream>>>(a32, a16, a8, wsBoxes, wsXyxy, wsLev, B);
  }
  // K1: sort + NMS + top-k per (level, batch)
  {
    dim3 grid(3, B);
    nms_topk_kernel<<<grid, 1024, 0, stream>>>(s32, s16, s8, wsBoxes, wsXyxy, wsLev,
                                               wsRois, wsValid);
  }
  // K2a: level p32 with WMMA pooling
  {
    dim3 grid(16, TOPK_N / ROI_CHUNK, B);
    size_t smem = (size_t)(16 * K2A_HW + 16 * K2A_SST + 16 * K2A_TST + 8 * 28) * 4;
    roi_align_lev0_kernel<<<grid, 256, smem, stream>>>(p32, wsRois, wsValid, out);
  }
  // K2b: levels p16/p8
  {
    dim3 grid(160, TOPK_N / ROI_CHUNK, B);
    size_t smem = (size_t)TILE_ELEMS * 4 + 8 * 28 * 4;
    roi_align_kernel<<<grid, K2_BLK, smem, stream>>>(p16, p8, wsRois, wsValid, out);
  }
}